// Causal_Temporal_Map_Attention_2_50362786513378
// MI455X (gfx1250) — compile-verified
//
#include <hip/hip_runtime.h>

// Problem sizes (compile-time constants from the reference)
#define Bc 16
#define Nc 2048
#define Tc 512
#define Ec 64
#define BM 64          // rows of N per workgroup (i-block) and per j-block
#define NTHREADS 512   // 16 wave32s
#define STRX 516       // padded LDS stride for 512-wide tiles (bank-conflict free, 8B aligned)
#define STRE 68        // padded LDS stride for 64-wide tiles

typedef float v2f __attribute__((ext_vector_type(2)));
typedef float v8f __attribute__((ext_vector_type(8)));

__device__ __forceinline__ v8f wmma4(v2f a, v2f b, v8f c) {
  // D = A(16x4,f32) * B(4x16,f32) + C(16x16,f32)
  return __builtin_amdgcn_wmma_f32_16x16x4_f32(false, a, false, b, (short)0, c,
                                               false, false);
}

// ---- gfx1250 async global->LDS copy (ASYNCcnt path, no VGPR round-trip) ----
__device__ __forceinline__ void async_ld_b64(float* lds_dst, const float* gsrc) {
  // VDST = LDS byte address (flat-shared addr[31:0] == LDS offset), VADDR = 64b global addr
  unsigned loff = (unsigned)(uintptr_t)(void*)lds_dst;
  asm volatile("global_load_async_to_lds_b64 %0, %1, off"
               :: "v"(loff), "v"(gsrc) : "memory");
}

__device__ __forceinline__ void async_wait_all() {
  asm volatile("s_wait_asynccnt 0" ::: "memory");
}

__device__ __forceinline__ void copy_tile_x(float* dst, const float* src, int tid) {
  // 64 x 512 f32, global row-major (ld=512) -> LDS (ld=STRX), async b64 per lane
  for (int t = tid; t < BM * (Tc / 2); t += NTHREADS) {
    int row = t >> 8;             // 256 float2 per row
    int c2  = (t & 255) * 2;
    async_ld_b64(&dst[row * STRX + c2], &src[(size_t)row * Tc + c2]);
  }
}

__device__ __forceinline__ void copy_tile_e(float* dst, const float* src, int tid) {
  // 64 x 64 f32 -> LDS (ld=STRE)
  for (int t = tid; t < BM * (Ec / 2); t += NTHREADS) {
    int row = t >> 5;             // 32 float2 per row
    int c2  = (t & 31) * 2;
    async_ld_b64(&dst[row * STRE + c2], &src[(size_t)row * Ec + c2]);
  }
}

__global__ __launch_bounds__(NTHREADS, 1)
void ctm_attn_fused(const float* __restrict__ x,  const float* __restrict__ e,
                    const float* __restrict__ W0, const float* __restrict__ W1,
                    const float* __restrict__ W2, const float* __restrict__ W3,
                    float* __restrict__ out) {
  extern __shared__ float smem[];
  float* Qs = smem;                  // 64 x STRX   (Q_i, persistent)
  float* Xs = Qs + BM * STRX;        // 64 x STRX   (x_i, then x_j per iter)
  float* Ps = Xs + BM * STRX;        // 64 x STRE   (P_i, persistent)
  float* Es = Ps + BM * STRE;        // 64 x STRE   (e_i, then e_j per iter)
  float* Ss = Es + BM * STRE;        // 64 x STRE   (S exchange tile)

  const int tid  = threadIdx.x;
  const int lane = tid & 31;
  const int w    = tid >> 5;     // wave id 0..15
  const int r    = lane & 15;    // lane row within half-wave
  const int h    = lane >> 4;    // half-wave select
  const int ti   = w & 3;        // 16-row tile index (all phases)
  const int cg   = w >> 2;       // column group (Q/out phases) == tj (S/P phases)

  const int ib = blockIdx.x;
  const int b  = blockIdx.y;
  const int i0 = ib * BM;
  const float* xb = x + (size_t)b * Nc * Tc;

  // ---------------- Phase 0: stage x_i, e_i (async -> LDS) ----------------
  copy_tile_x(Xs, xb + (size_t)i0 * Tc, tid);
  copy_tile_e(Es, e + (size_t)i0 * Ec, tid);
  async_wait_all();
  __syncthreads();

  // ---------------- Phase 1a: Q_i = x_i @ relu(triu(W0,1))^T + e_i @ W2^T ----
  {
    v8f q[8];
    #pragma unroll
    for (int tl = 0; tl < 8; ++tl) q[tl] = (v8f){0.f,0.f,0.f,0.f,0.f,0.f,0.f,0.f};

    const int arow = (16 * ti + r) * STRX;
    for (int kb = 0; kb < Tc; kb += 4) {
      v2f a = *(const v2f*)&Xs[arow + kb + 2 * h];
      const int k0 = kb + 2 * h;
      #pragma unroll
      for (int tl = 0; tl < 8; ++tl) {
        const int n = cg * 128 + tl * 16 + r;          // output column of Q = row of W0
        // unconditional vector load, then branchless relu(triu(W0,1)) mask
        v2f wv = *(const v2f*)&W0[(size_t)n * Tc + k0];
        v2f bf;
        bf.x = (k0     > n) ? fmaxf(wv.x, 0.f) : 0.f;
        bf.y = (k0 + 1 > n) ? fmaxf(wv.y, 0.f) : 0.f;
        q[tl] = wmma4(a, bf, q[tl]);
      }
    }
    const int erow = (16 * ti + r) * STRE;
    for (int kb = 0; kb < Ec; kb += 4) {
      v2f a = *(const v2f*)&Es[erow + kb + 2 * h];
      const int k0 = kb + 2 * h;
      #pragma unroll
      for (int tl = 0; tl < 8; ++tl) {
        const int n = cg * 128 + tl * 16 + r;
        v2f bf = *(const v2f*)&W2[(size_t)n * Ec + k0];
        q[tl] = wmma4(a, bf, q[tl]);
      }
    }
    #pragma unroll
    for (int tl = 0; tl < 8; ++tl)
      #pragma unroll
      for (int v = 0; v < 8; ++v)
        Qs[(16 * ti + v + 8 * h) * STRX + cg * 128 + tl * 16 + r] = q[tl][v];
  }

  // ---------------- Phase 1b: P_i = x_i @ W1^T + e_i @ W3^T ------------------
  {
    v8f p = (v8f){0.f,0.f,0.f,0.f,0.f,0.f,0.f,0.f};
    const int tj = cg;
    const int arow = (16 * ti + r) * STRX;
    for (int kb = 0; kb < Tc; kb += 4) {
      v2f a = *(const v2f*)&Xs[arow + kb + 2 * h];
      const int k0 = kb + 2 * h;
      const int n  = 16 * tj + r;
      v2f bf = *(const v2f*)&W1[(size_t)n * Tc + k0];
      p = wmma4(a, bf, p);
    }
    const int erow = (16 * ti + r) * STRE;
    for (int kb = 0; kb < Ec; kb += 4) {
      v2f a = *(const v2f*)&Es[erow + kb + 2 * h];
      const int k0 = kb + 2 * h;
      const int n  = 16 * tj + r;
      v2f bf = *(const v2f*)&W3[(size_t)n * Ec + k0];
      p = wmma4(a, bf, p);
    }
    #pragma unroll
    for (int v = 0; v < 8; ++v)
      Ps[(16 * ti + v + 8 * h) * STRE + 16 * tj + r] = p[v];
  }

  // ---------------- Phase 2: flash loop over j-blocks ------------------------
  v8f o[8];
  #pragma unroll
  for (int tl = 0; tl < 8; ++tl) o[tl] = (v8f){0.f,0.f,0.f,0.f,0.f,0.f,0.f,0.f};

  for (int jb = 0; jb < Nc / BM; ++jb) {
    const int j0 = jb * BM;
    __syncthreads();   // Qs/Ps stores + previous out-phase done before Xs/Es reuse

    if (jb + 1 < Nc / BM) {   // prefetch next x_j block (global_prefetch_b8)
      const float* pf = xb + (size_t)(j0 + BM + (tid >> 3)) * Tc + (tid & 7) * 64;
      __builtin_prefetch(pf, 0, 0);
    }

    copy_tile_x(Xs, xb + (size_t)j0 * Tc, tid);
    copy_tile_e(Es, e + (size_t)j0 * Ec, tid);
    async_wait_all();
    __syncthreads();

    // S = Q_i @ x_j^T + P_i @ e_j^T  (each wave: one 16x16 tile, K=512+64)
    {
      v8f s = (v8f){0.f,0.f,0.f,0.f,0.f,0.f,0.f,0.f};
      const int tj   = cg;
      const int qrow = (16 * ti + r) * STRX;
      const int xrow = (16 * tj + r) * STRX;   // B^T: B[k=t][n=m] = x_j[m][t]
      #pragma unroll 4
      for (int kb = 0; kb < Tc; kb += 4) {
        v2f a  = *(const v2f*)&Qs[qrow + kb + 2 * h];
        v2f bf = *(const v2f*)&Xs[xrow + kb + 2 * h];
        s = wmma4(a, bf, s);
      }
      const int prow = (16 * ti + r) * STRE;
      const int erow = (16 * tj + r) * STRE;
      #pragma unroll 4
      for (int kb = 0; kb < Ec; kb += 4) {
        v2f a  = *(const v2f*)&Ps[prow + kb + 2 * h];
        v2f bf = *(const v2f*)&Es[erow + kb + 2 * h];
        s = wmma4(a, bf, s);
      }
      #pragma unroll
      for (int v = 0; v < 8; ++v)
        Ss[(16 * ti + v + 8 * h) * STRE + 16 * tj + r] = s[v];
    }
    __syncthreads();

    // out += S @ x_j  (each wave: rows 16*ti..+16, cols cg*128..+128, K=64)
    {
      v2f afr[16];
      const int srow = (16 * ti + r) * STRE;
      #pragma unroll
      for (int ks = 0; ks < 16; ++ks)
        afr[ks] = *(const v2f*)&Ss[srow + 4 * ks + 2 * h];
      #pragma unroll
      for (int tl = 0; tl < 8; ++tl) {
        const int col = cg * 128 + tl * 16 + r;
        #pragma unroll
        for (int ks = 0; ks < 16; ++ks) {
          const int k0 = 4 * ks + 2 * h;
          v2f bf = { Xs[k0 * STRX + col], Xs[(k0 + 1) * STRX + col] };
          o[tl] = wmma4(afr[ks], bf, o[tl]);
        }
      }
    }
  }

  // ---------------- Epilogue: write out[b, i0..i0+64, 0..512] ----------------
  #pragma unroll
  for (int tl = 0; tl < 8; ++tl)
    #pragma unroll
    for (int v = 0; v < 8; ++v)
      out[((size_t)b * Nc + i0 + 16 * ti + v + 8 * h) * Tc + cg * 128 + tl * 16 + r] =
          o[tl][v];
}

extern "C" void kernel_launch(void* const* d_in, const int* in_sizes, int n_in,
                              void* d_out, int out_size, void* d_ws, size_t ws_size,
                              hipStream_t stream) {
  (void)in_sizes; (void)n_in; (void)out_size; (void)d_ws; (void)ws_size;
  const float* x  = (const float*)d_in[0];
  const float* e  = (const float*)d_in[1];
  const float* W0 = (const float*)d_in[2];
  const float* W1 = (const float*)d_in[3];
  const float* W2 = (const float*)d_in[4];
  const float* W3 = (const float*)d_in[5];
  float* out = (float*)d_out;

  const size_t shmem =
      (size_t)(2 * BM * STRX + 3 * BM * STRE) * sizeof(float);  // 316416 B <= 320KB WGP LDS

  // Opt in to large dynamic LDS (not a stream op; safe under graph capture).
  (void)hipFuncSetAttribute((const void*)ctm_attn_fused,
                            hipFuncAttributeMaxDynamicSharedMemorySize,
                            (int)shmem);

  dim3 grid(Nc / BM, Bc);
  dim3 block(NTHREADS);
  hipLaunchKernelGGL(ctm_attn_fused, grid, block, shmem, stream,
                     x, e, W0, W1, W2, W3, out);
}